// QuantumValueNetwork_3564822856062
// MI455X (gfx1250) — compile-verified
//
#include <hip/hip_runtime.h>

// MI455X / gfx1250, wave32. One batch element per wave; 256-dim complex64
// state kept entirely in VGPRs as a 16x16 complex matrix in WMMA C/D layout:
//   VGPR r, lane<16  -> S[r,     lane]      (row = idx[7:4], col = idx[3:0])
//   VGPR r, lane>=16 -> S[r + 8, lane - 16]
// Qubits 0..3 (row bits) -> fused 16x16 gate, applied with V_WMMA_F32_16X16X4_F32.
// Qubits 4..7 (col bits = lane[3:0]) -> 2x2 gates via shfl_xor partner exchange.

typedef float v2f __attribute__((ext_vector_type(2)));
typedef float v8f __attribute__((ext_vector_type(8)));

#define NQ 8
#define NL 3
#define OBS 11

__device__ __forceinline__ float sx(float v, int m) { return __shfl_xor(v, m, 32); }

__global__ __launch_bounds__(256) void qvn_kernel(
    const float* __restrict__ s, const float* __restrict__ mu,
    const float* __restrict__ sigma, const float* __restrict__ theta,
    const float* __restrict__ w, const float* __restrict__ aa,
    const float* __restrict__ bb, float* __restrict__ out, int batch)
{
  const int lane = threadIdx.x & 31;
  const int wid  = (blockIdx.x * blockDim.x + threadIdx.x) >> 5;
  if (wid >= batch) return;            // wave-uniform branch, EXEC stays full

  const int hi  = lane >> 4;           // row bit 3 (qubit 0)
  const int col = lane & 15;           // idx[3:0]  (qubits 4..7)

  // observation angles (tiny uniform loads, every lane loads its copy)
  float x[NQ];
  #pragma unroll
  for (int q = 0; q < NQ; ++q)
    x[q] = atanf((s[wid * OBS + q] - mu[q]) / (sigma[q] + 1e-8f));

  // CZ-entangler signs for this lane's 8 state elements:
  // sign = (-1)^(# adjacent 1-bit pairs of idx)
  float esgn[8];
  #pragma unroll
  for (int r = 0; r < 8; ++r) {
    int idx = ((r + 8 * hi) << 4) | col;
    int t   = idx & (idx >> 1) & 0x7F;
    esgn[r] = (__popc(t) & 1) ? -1.0f : 1.0f;
  }

  // |0...0>  (pre-CZ sign at index 0 is +1)
  float sr[8], si[8];
  #pragma unroll
  for (int r = 0; r < 8; ++r) { sr[r] = 0.0f; si[r] = 0.0f; }
  sr[0] = (lane == 0) ? 1.0f : 0.0f;

  for (int l = 0; l < NL; ++l) {
    // ---- per-qubit Rot(phi,theta,omega) 2x2 entries ----
    float g00r[NQ], g00i[NQ], g01r[NQ], g01i[NQ],
          g10r[NQ], g10i[NQ], g11r[NQ], g11i[NQ];
    #pragma unroll
    for (int q = 0; q < NQ; ++q) {
      const int base = (l * NQ + q) * 3;
      float phi = theta[base + 0] + w[base + 0] * x[q];
      float th  = theta[base + 1] + w[base + 1] * x[q];
      float om  = theta[base + 2] + w[base + 2] * x[q];
      float cth, sth, ca, sa, cb, sb;
      __sincosf(0.5f * th,         &sth, &cth);
      __sincosf(0.5f * (phi + om), &sa,  &ca);
      __sincosf(0.5f * (phi - om), &sb,  &cb);
      g00r[q] =  cth * ca;  g00i[q] = -cth * sa;   // e^{-i(phi+om)/2} c
      g01r[q] = -sth * cb;  g01i[q] = -sth * sb;   // -e^{ i(phi-om)/2} s
      g10r[q] =  sth * cb;  g10i[q] = -sth * sb;   //  e^{-i(phi-om)/2} s
      g11r[q] =  cth * ca;  g11i[q] =  cth * sa;   // e^{ i(phi+om)/2} c
    }

    // ---- fused 16x16 gate G = g0 (x) g1 (x) g2 (x) g3, built directly in
    //      WMMA A layout: reg 2c -> K = 4c + 2*hi, reg 2c+1 -> K = 4c + 2*hi + 1,
    //      M = lane % 16.  AiN = -Ai (float WMMA has no A/B negate bits). ----
    float Ar[8], Ai[8], AiN[8];
    #pragma unroll
    for (int j = 0; j < 8; ++j) {
      int k = 4 * (j >> 1) + 2 * hi + (j & 1);
      float pr = 1.0f, pi = 0.0f;
      #pragma unroll
      for (int q = 0; q < 4; ++q) {
        int mb = (col >> (3 - q)) & 1;
        int kb = (k   >> (3 - q)) & 1;
        float er = mb ? (kb ? g11r[q] : g10r[q]) : (kb ? g01r[q] : g00r[q]);
        float ei = mb ? (kb ? g11i[q] : g10i[q]) : (kb ? g01i[q] : g00i[q]);
        float nr = pr * er - pi * ei;
        float ni = pr * ei + pi * er;
        pr = nr; pi = ni;
      }
      Ar[j] = pr; Ai[j] = pi; AiN[j] = -pi;
    }

    // ---- state rows -> WMMA B layout (K halves split across lane halves) ----
    float t0r = sx(sr[0],16), t1r = sx(sr[1],16), t2r = sx(sr[2],16), t3r = sx(sr[3],16);
    float t4r = sx(sr[4],16), t5r = sx(sr[5],16), t6r = sx(sr[6],16), t7r = sx(sr[7],16);
    float t0i = sx(si[0],16), t1i = sx(si[1],16), t2i = sx(si[2],16), t3i = sx(si[3],16);
    float t4i = sx(si[4],16), t5i = sx(si[5],16), t6i = sx(si[6],16), t7i = sx(si[7],16);
    const bool lo = (hi == 0);
    float Br[8], Bi[8];
    Br[0] = lo ? sr[0] : t2r;  Bi[0] = lo ? si[0] : t2i;   // K = 0 / 2
    Br[1] = lo ? sr[1] : t3r;  Bi[1] = lo ? si[1] : t3i;   // K = 1 / 3
    Br[2] = lo ? sr[4] : t6r;  Bi[2] = lo ? si[4] : t6i;   // K = 4 / 6
    Br[3] = lo ? sr[5] : t7r;  Bi[3] = lo ? si[5] : t7i;   // K = 5 / 7
    Br[4] = lo ? t0r : sr[2];  Bi[4] = lo ? t0i : si[2];   // K = 8 / 10
    Br[5] = lo ? t1r : sr[3];  Bi[5] = lo ? t1i : si[3];   // K = 9 / 11
    Br[6] = lo ? t4r : sr[6];  Bi[6] = lo ? t4i : si[6];   // K = 12 / 14
    Br[7] = lo ? t5r : sr[7];  Bi[7] = lo ? t5i : si[7];   // K = 13 / 15

    // ---- complex 16x16x16 left-multiply: 16x V_WMMA_F32_16X16X4_F32 ----
    v8f cr  = {0.f,0.f,0.f,0.f,0.f,0.f,0.f,0.f};
    v8f ci_ = {0.f,0.f,0.f,0.f,0.f,0.f,0.f,0.f};
    #pragma unroll
    for (int c = 0; c < 4; ++c) {
      v2f a_r = { Ar[2*c],  Ar[2*c+1]  };
      v2f a_i = { Ai[2*c],  Ai[2*c+1]  };
      v2f a_n = { AiN[2*c], AiN[2*c+1] };
      v2f b_r = { Br[2*c],  Br[2*c+1]  };
      v2f b_i = { Bi[2*c],  Bi[2*c+1]  };
      cr  = __builtin_amdgcn_wmma_f32_16x16x4_f32(false, a_r, false, b_r, (short)0, cr,  false, false);
      cr  = __builtin_amdgcn_wmma_f32_16x16x4_f32(false, a_n, false, b_i, (short)0, cr,  false, false);
      ci_ = __builtin_amdgcn_wmma_f32_16x16x4_f32(false, a_r, false, b_i, (short)0, ci_, false, false);
      ci_ = __builtin_amdgcn_wmma_f32_16x16x4_f32(false, a_i, false, b_r, (short)0, ci_, false, false);
    }
    #pragma unroll
    for (int r = 0; r < 8; ++r) { sr[r] = cr[r]; si[r] = ci_[r]; }

    // ---- qubits 4..7: 2x2 gates on column bits via lane exchange ----
    #pragma unroll
    for (int q = 4; q < 8; ++q) {
      int b    = 7 - q;               // lane bit 3..0
      int mask = 1 << b;
      int cb   = (lane >> b) & 1;
      float c0r = cb ? g11r[q] : g00r[q];   // self coefficient
      float c0i = cb ? g11i[q] : g00i[q];
      float c1r = cb ? g10r[q] : g01r[q];   // partner coefficient
      float c1i = cb ? g10i[q] : g01i[q];
      #pragma unroll
      for (int r = 0; r < 8; ++r) {
        float pr = sx(sr[r], mask);
        float pi = sx(si[r], mask);
        float nr = c0r * sr[r] - c0i * si[r] + c1r * pr - c1i * pi;
        float ni = c0r * si[r] + c0i * sr[r] + c1r * pi + c1i * pr;
        sr[r] = nr; si[r] = ni;
      }
    }

    // ---- CZ entangler diagonal ----
    #pragma unroll
    for (int r = 0; r < 8; ++r) { sr[r] *= esgn[r]; si[r] *= esgn[r]; }
  }

  // ---- <Z> on qubit 0: idx bit7 == (lane >= 16) ----
  float p = 0.0f;
  #pragma unroll
  for (int r = 0; r < 8; ++r) p += sr[r] * sr[r] + si[r] * si[r];
  p = hi ? -p : p;
  #pragma unroll
  for (int m = 16; m >= 1; m >>= 1) p += sx(p, m);
  if (lane == 0) out[wid] = aa[0] * p + bb[0];
}

extern "C" void kernel_launch(void* const* d_in, const int* in_sizes, int n_in,
                              void* d_out, int out_size, void* d_ws, size_t ws_size,
                              hipStream_t stream) {
  (void)n_in; (void)d_ws; (void)ws_size; (void)out_size;
  const float* s     = (const float*)d_in[0];
  const float* mu    = (const float*)d_in[1];
  const float* sigma = (const float*)d_in[2];
  const float* theta = (const float*)d_in[3];
  const float* w     = (const float*)d_in[4];
  const float* a     = (const float*)d_in[5];
  const float* b     = (const float*)d_in[6];
  float* out = (float*)d_out;

  const int batch = in_sizes[0] / OBS;          // 32768
  const int threads = 256;                      // 8 waves / block
  const int waves_per_block = threads / 32;
  const int grid = (batch + waves_per_block - 1) / waves_per_block;
  qvn_kernel<<<grid, threads, 0, stream>>>(s, mu, sigma, theta, w, a, b, out, batch);
}